// RRNet_PointerAttention_82291573391771
// MI455X (gfx1250) — compile-verified
//
#include <hip/hip_runtime.h>

typedef __attribute__((ext_vector_type(2))) float v2f;
typedef __attribute__((ext_vector_type(8))) float v8f;

constexpr int B  = 256;
constexpr int S  = 1000;
constexpr int E  = 512;
constexpr int H  = 8;
constexpr int DH = 64;
constexpr int DFF = 2048;

// ---------------------------------------------------------------------------
// gfx1250 async global->LDS DMA (ASYNCcnt path), per CDNA5 ISA
// Writes 16 bytes from global memory directly into LDS, no VGPR round trip.
// ---------------------------------------------------------------------------
__device__ __forceinline__ void async_load_b128_to_lds(void* lds, const void* gptr) {
  unsigned lds_addr = (unsigned)(unsigned long long)lds;  // low 32 bits = LDS offset
  asm volatile("global_load_async_to_lds_b128 %0, %1, off"
               :: "v"(lds_addr), "v"(gptr)
               : "memory");
}
__device__ __forceinline__ void wait_async0() {
  asm volatile("s_wait_asynccnt 0x0" ::: "memory");
}

// ---------------------------------------------------------------------------
// Kernel 1: masked multi-head attention (L=1) + residual  ->  G = heads + q
// One block per batch. Streams K and V exactly once (coalesced).
// ---------------------------------------------------------------------------
__global__ __launch_bounds__(512) void attn_kernel(
    const float* __restrict__ q, const float* __restrict__ kmat,
    const float* __restrict__ vmat, const int* __restrict__ mask,
    float* __restrict__ G)
{
  __shared__ float4 qs4[E / 4];
  __shared__ float  sc[H][S + 8];
  __shared__ float  invs[H];

  const int b   = blockIdx.x;
  const int tid = threadIdx.x;

  if (tid < E / 4) qs4[tid] = ((const float4*)(q + (size_t)b * E))[tid];
  __syncthreads();

  // ---- scores: sc[h][s] = (q_h . k_h[s]) / sqrt(DH), masked ----
  for (int s = tid; s < S; s += blockDim.x) {
    const float4* krow = (const float4*)(kmat + ((size_t)b * S + s) * E);
    const int mk = mask[(size_t)b * S + s];
#pragma unroll
    for (int h = 0; h < H; h++) {
      float a = 0.f;
#pragma unroll
      for (int j = 0; j < DH / 4; j++) {
        float4 kv = krow[h * (DH / 4) + j];
        float4 qv = qs4[h * (DH / 4) + j];
        a = fmaf(kv.x, qv.x, a); a = fmaf(kv.y, qv.y, a);
        a = fmaf(kv.z, qv.z, a); a = fmaf(kv.w, qv.w, a);
      }
      sc[h][s] = mk ? a * 0.125f : -__builtin_inff();
    }
  }
  __syncthreads();

  // ---- per-head softmax: wave w reduces head w ----
  const int wave = tid >> 5, lane = tid & 31;
  if (wave < H) {
    float m = -__builtin_inff();
    for (int s = lane; s < S; s += 32) m = fmaxf(m, sc[wave][s]);
#pragma unroll
    for (int off = 16; off > 0; off >>= 1) m = fmaxf(m, __shfl_xor(m, off, 32));
    float sum = 0.f;
    for (int s = lane; s < S; s += 32) {
      float e = __expf(sc[wave][s] - m);
      sc[wave][s] = e;
      sum += e;
    }
#pragma unroll
    for (int off = 16; off > 0; off >>= 1) sum += __shfl_xor(sum, off, 32);
    if (lane == 0) invs[wave] = 1.f / sum;
  }
  __syncthreads();

  // ---- heads = p . V (column-coalesced), + residual ----
  const int e = tid;           // 512 threads == E outputs
  const int h = e >> 6;
  const float* vcol = vmat + (size_t)b * S * E + e;
  const float* p = &sc[h][0];
  float acc = 0.f;
#pragma unroll 4
  for (int s = 0; s < S; s++) acc = fmaf(p[s], vcol[(size_t)s * E], acc);
  G[(size_t)b * E + e] = acc * invs[h] + ((const float*)qs4)[e];
}

// ---------------------------------------------------------------------------
// Kernel 2a: Hid = relu(G @ W1 + b1)   [256x512]x[512x2048], fp32 WMMA
// grid (16,16): blockIdx.x = M tile, blockIdx.y = 128-col N group. 8 waves.
// A tile staged into LDS with async global->LDS DMA.
// ---------------------------------------------------------------------------
__global__ __launch_bounds__(256) void ffn1_kernel(
    const float* __restrict__ G, const float* __restrict__ W1,
    const float* __restrict__ b1, float* __restrict__ Hid)
{
  __shared__ float lA[16][E + 4];              // row stride 2064B (16B aligned)
  const int tid  = threadIdx.x;
  const int wave = tid >> 5, lane = tid & 31;
  const int m0 = blockIdx.x * 16;
  const int n  = blockIdx.y * 128 + wave * 16 + (lane & 15);

  for (int i = tid; i < 16 * (E / 4); i += 256) {
    int r = i / (E / 4), c = (i % (E / 4)) * 4;
    async_load_b128_to_lds(&lA[r][c], &G[(size_t)(m0 + r) * E + c]);
  }
  wait_async0();
  __syncthreads();

  const int m = lane & 15;
  const int khalf = (lane >> 4) << 1;          // 0 for lanes 0-15, 2 for 16-31
  v8f acc = {};
  for (int k = 0; k < E; k += 4) {
    v2f a; a.x = lA[m][k + khalf]; a.y = lA[m][k + khalf + 1];
    const float* bp = W1 + (size_t)(k + khalf) * DFF + n;
    v2f bb; bb.x = bp[0]; bb.y = bp[DFF];
    acc = __builtin_amdgcn_wmma_f32_16x16x4_f32(false, a, false, bb,
                                                (short)0, acc, false, false);
  }

  const int mb = m0 + ((lane >> 4) << 3);      // rows r / r+8 per lane half
  const float bias = b1[n];
#pragma unroll
  for (int r = 0; r < 8; r++)
    Hid[(size_t)(mb + r) * DFF + n] = fmaxf(acc[r] + bias, 0.f);
}

// ---------------------------------------------------------------------------
// Kernel 2b: G2 = Hid @ W2 + b2 + G   [256x2048]x[2048x512], fp32 WMMA
// grid (16,4). K=2048 chunked through LDS in 512-wide slabs (async DMA).
// ---------------------------------------------------------------------------
__global__ __launch_bounds__(256) void ffn2_kernel(
    const float* __restrict__ Hid, const float* __restrict__ W2,
    const float* __restrict__ b2, const float* __restrict__ G,
    float* __restrict__ G2)
{
  __shared__ float lA[16][512 + 4];
  const int tid  = threadIdx.x;
  const int wave = tid >> 5, lane = tid & 31;
  const int m0 = blockIdx.x * 16;
  const int n  = blockIdx.y * 128 + wave * 16 + (lane & 15);
  const int m = lane & 15;
  const int khalf = (lane >> 4) << 1;

  v8f acc = {};
  for (int kc = 0; kc < DFF; kc += 512) {
    for (int i = tid; i < 16 * 128; i += 256) {
      int r = i >> 7, c = (i & 127) * 4;
      async_load_b128_to_lds(&lA[r][c], &Hid[(size_t)(m0 + r) * DFF + kc + c]);
    }
    wait_async0();
    __syncthreads();
    for (int k = 0; k < 512; k += 4) {
      v2f a; a.x = lA[m][k + khalf]; a.y = lA[m][k + khalf + 1];
      const float* bp = W2 + (size_t)(kc + k + khalf) * E + n;
      v2f bb; bb.x = bp[0]; bb.y = bp[E];
      acc = __builtin_amdgcn_wmma_f32_16x16x4_f32(false, a, false, bb,
                                                  (short)0, acc, false, false);
    }
    __syncthreads();
  }

  const int mb = m0 + ((lane >> 4) << 3);
  const float bias = b2[n];
#pragma unroll
  for (int r = 0; r < 8; r++)
    G2[(size_t)(mb + r) * E + n] =
        acc[r] + bias + G[(size_t)(mb + r) * E + n];
}

// ---------------------------------------------------------------------------
// Kernel 4: logits[b][s] = (G2[b] . logit_key[b][s]) / sqrt(E)
// One block per batch, streams logit_key once.
// ---------------------------------------------------------------------------
__global__ __launch_bounds__(512) void logits_kernel(
    const float* __restrict__ G2, const float* __restrict__ LK,
    float* __restrict__ out)
{
  __shared__ float4 gs[E / 4];
  const int b = blockIdx.x, tid = threadIdx.x;
  if (tid < E / 4) gs[tid] = ((const float4*)(G2 + (size_t)b * E))[tid];
  __syncthreads();

  const float scale = 0.04419417382415922f;    // 1/sqrt(512)
  for (int s = tid; s < S; s += blockDim.x) {
    const float4* lkr = (const float4*)(LK + ((size_t)b * S + s) * E);
    float a = 0.f;
#pragma unroll 8
    for (int j = 0; j < E / 4; j++) {
      float4 x = lkr[j]; float4 g = gs[j];
      a = fmaf(x.x, g.x, a); a = fmaf(x.y, g.y, a);
      a = fmaf(x.z, g.z, a); a = fmaf(x.w, g.w, a);
    }
    out[(size_t)b * S + s] = a * scale;
  }
}

// ---------------------------------------------------------------------------
extern "C" void kernel_launch(void* const* d_in, const int* in_sizes, int n_in,
                              void* d_out, int out_size, void* d_ws, size_t ws_size,
                              hipStream_t stream) {
  (void)in_sizes; (void)n_in; (void)out_size; (void)ws_size;
  const float* q    = (const float*)d_in[0];
  const float* kmat = (const float*)d_in[1];
  const float* vmat = (const float*)d_in[2];
  const float* lk   = (const float*)d_in[3];
  const int*   mask = (const int*)  d_in[4];
  const float* W1   = (const float*)d_in[5];
  const float* b1   = (const float*)d_in[6];
  const float* W2   = (const float*)d_in[7];
  const float* b2   = (const float*)d_in[8];
  float* out = (float*)d_out;

  float* G   = (float*)d_ws;                 // [256 x 512]
  float* Hid = G + (size_t)B * E;            // [256 x 2048]
  float* G2  = Hid + (size_t)B * DFF;        // [256 x 512]

  attn_kernel  <<<B, 512, 0, stream>>>(q, kmat, vmat, mask, G);
  ffn1_kernel  <<<dim3(B / 16, DFF / 128), 256, 0, stream>>>(G, W1, b1, Hid);
  ffn2_kernel  <<<dim3(B / 16, E / 128),   256, 0, stream>>>(Hid, W2, b2, G, G2);
  logits_kernel<<<B, 512, 0, stream>>>(G2, lk, out);
}